// HMMNet_82274393523067
// MI455X (gfx1250) — compile-verified
//
#include <hip/hip_runtime.h>
#include <stdint.h>

// HMM forward scan, log-space, rank-1+diag transition structure exploited:
//   f_new[j] = al[t,j] + logaddexp(start[j] + LSE_i(f[i]+beta[i]), f[j] + omb[j])
// Phase 1: parallel gather al[t,j] = action_logps[t,j,actions[t]] -> workspace.
// Phase 2: single wave32 scan, 4 states/lane, TDM double-buffered LDS prefetch.

#define NB 128   // states (b)
#define NA 256   // actions (a)
#define CH 16    // rows per TDM chunk

typedef unsigned int v4u __attribute__((ext_vector_type(4)));
typedef int          v8i __attribute__((ext_vector_type(8)));
typedef int          v4i __attribute__((ext_vector_type(4)));

// ---- Tensor Data Mover: 2D tile (tile_rows x row_elems f32) global -> LDS ----
__device__ __forceinline__ void tdm_load_2d(const float* gptr, unsigned lds_addr,
                                            unsigned row_elems, unsigned tile_rows,
                                            unsigned row_stride_elems)
{
    unsigned long long ga = (unsigned long long)(uintptr_t)gptr;
    const unsigned tdim0 = row_elems;
    const unsigned tdim1 = 1u << 20;   // generous bound: no OOB clipping needed

    v4u g0;
    g0.x = 1u;                                       // count=1 (valid), no gather
    g0.y = lds_addr;                                 // [63:32] LDS byte address
    g0.z = (unsigned)(ga & 0xFFFFFFFFull);           // [95:64] global_addr lo
    g0.w = (unsigned)((ga >> 32) & 0x1FFFFFFull)     // [120:96] global_addr hi
         | (2u << 30);                               // [127:126] type=2 (image)

    v8i g1;
    g1[0] = (int)(2u << 16);                                   // data_size=4B; wg_mask=0
    g1[1] = (int)((tdim0 & 0xFFFFu) << 16);                    // tensor_dim0[15:0]
    g1[2] = (int)((tdim0 >> 16) & 0xFFFFu)
          | (int)((tdim1 & 0xFFFFu) << 16);                    // dim0 hi | dim1 lo
    g1[3] = (int)((tdim1 >> 16) & 0xFFFFu)
          | (int)((row_elems & 0xFFFFu) << 16);                // dim1 hi | tile_dim0
    g1[4] = (int)(tile_rows & 0xFFFFu);                        // tile_dim1 (tile_dim2=0)
    g1[5] = (int)row_stride_elems;                             // tensor_dim0_stride lo
    g1[6] = 0;                                                 // stride hi | dim1_stride lo
    g1[7] = 0;

    v4i z4 = {0, 0, 0, 0};
#if __clang_major__ >= 23
    v8i z8 = {0, 0, 0, 0, 0, 0, 0, 0};
    __builtin_amdgcn_tensor_load_to_lds(g0, g1, z4, z4, z8, 0);
#else
    __builtin_amdgcn_tensor_load_to_lds(g0, g1, z4, z4, 0);
#endif
}

// ---- wave32 reductions ----
__device__ __forceinline__ float wave_max(float v) {
#pragma unroll
    for (int o = 16; o > 0; o >>= 1) v = fmaxf(v, __shfl_xor(v, o, 32));
    return v;
}
__device__ __forceinline__ float wave_sum(float v) {
#pragma unroll
    for (int o = 16; o > 0; o >>= 1) v += __shfl_xor(v, o, 32);
    return v;
}
__device__ __forceinline__ float lae(float x, float y) {   // logaddexp
    float m = fmaxf(x, y);
    return m + __logf(__expf(x - m) + __expf(y - m));
}

// ---- Phase 1: densify al[t,j] = action_logps[t, j, actions[t]] ----
__global__ void gather_al(const float* __restrict__ action_logps,
                          const int* __restrict__ actions,
                          float* __restrict__ al, int T)
{
    int idx = blockIdx.x * blockDim.x + threadIdx.x;
    if (idx >= T * NB) return;
    int t = idx >> 7;            // / NB
    int j = idx & (NB - 1);
    int act = actions[t];
    al[idx] = action_logps[(size_t)t * (NB * NA) + (size_t)j * NA + (size_t)act];
}

// ---- Phase 2: single-wave sequential scan with TDM double-buffering ----
__global__ void __launch_bounds__(32) hmm_scan(
    const float* __restrict__ stop_logps,   // (T+1, 128, 2): [beta, omb] pairs
    const float* __restrict__ start_logps,  // (T+1, 128)
    const float* __restrict__ al,           // (T, 128) densified
    float* __restrict__ out, int T)
{
    __shared__ alignas(16) float s_al[2][CH * NB];
    __shared__ alignas(16) float s_st[2][CH * NB];
    __shared__ alignas(16) float s_sp[2][CH * NB * 2];

    const int lane = threadIdx.x;   // lane holds states j = 4*lane .. 4*lane+3

    // f0 = start_logps[0] + al[0]
    float4 f;
    {
        const float4 s0 = *(const float4*)&start_logps[4 * lane];
        const float4 a0 = *(const float4*)&al[4 * lane];
        f.x = s0.x + a0.x; f.y = s0.y + a0.y; f.z = s0.z + a0.z; f.w = s0.w + a0.w;
    }

    const int steps = T - 1;                    // t = 1 .. T-1
    const int nch   = (steps + CH - 1) / CH;

    auto issue = [&](int c, int buf) {
        const int tbase = 1 + c * CH;
        int rows = T - tbase; if (rows > CH) rows = CH;
        tdm_load_2d(al          + (size_t)tbase * NB,     (unsigned)(uintptr_t)&s_al[buf][0], NB,     (unsigned)rows, NB);
        tdm_load_2d(start_logps + (size_t)tbase * NB,     (unsigned)(uintptr_t)&s_st[buf][0], NB,     (unsigned)rows, NB);
        tdm_load_2d(stop_logps  + (size_t)tbase * NB * 2, (unsigned)(uintptr_t)&s_sp[buf][0], NB * 2, (unsigned)rows, NB * 2);
    };

    issue(0, 0);
    for (int c = 0; c < nch; ++c) {
        const int buf = c & 1;
        if (c + 1 < nch) {
            issue(c + 1, buf ^ 1);
            __builtin_amdgcn_s_wait_tensorcnt(3);   // chunk c's 3 loads complete
        } else {
            __builtin_amdgcn_s_wait_tensorcnt(0);
        }

        const int tbase = 1 + c * CH;
        int rows = T - tbase; if (rows > CH) rows = CH;

        for (int r = 0; r < rows; ++r) {
            const float4 a4 = *(const float4*)&s_al[buf][r * NB + 4 * lane];
            const float4 t4 = *(const float4*)&s_st[buf][r * NB + 4 * lane];
            const float4 p0 = *(const float4*)&s_sp[buf][r * NB * 2 + 8 * lane];
            const float4 p1 = *(const float4*)&s_sp[buf][r * NB * 2 + 8 * lane + 4];
            // beta = {p0.x, p0.z, p1.x, p1.z}, omb = {p0.y, p0.w, p1.y, p1.w}

            // L = LSE_i(f[i] + beta[i]) across all 128 states
            const float x0 = f.x + p0.x, x1 = f.y + p0.z;
            const float x2 = f.z + p1.x, x3 = f.w + p1.z;
            float m = wave_max(fmaxf(fmaxf(x0, x1), fmaxf(x2, x3)));
            float s = wave_sum(__expf(x0 - m) + __expf(x1 - m) +
                               __expf(x2 - m) + __expf(x3 - m));
            const float L = m + __logf(s);

            // f[j] = al[t,j] + logaddexp(start[j] + L, f[j] + omb[j])
            f.x = a4.x + lae(t4.x + L, f.x + p0.y);
            f.y = a4.y + lae(t4.y + L, f.y + p0.w);
            f.z = a4.z + lae(t4.z + L, f.z + p1.y);
            f.w = a4.w + lae(t4.w + L, f.w + p1.w);
        }
    }

    // total = LSE_j(f[j] + stop_logps[T, j, STOP]);  out = -total
    const float4 q0 = *(const float4*)&stop_logps[(size_t)T * NB * 2 + 8 * lane];
    const float4 q1 = *(const float4*)&stop_logps[(size_t)T * NB * 2 + 8 * lane + 4];
    const float y0 = f.x + q0.x, y1 = f.y + q0.z;
    const float y2 = f.z + q1.x, y3 = f.w + q1.z;
    float m = wave_max(fmaxf(fmaxf(y0, y1), fmaxf(y2, y3)));
    float s = wave_sum(__expf(y0 - m) + __expf(y1 - m) +
                       __expf(y2 - m) + __expf(y3 - m));
    if (lane == 0) out[0] = -(m + __logf(s));
}

extern "C" void kernel_launch(void* const* d_in, const int* in_sizes, int n_in,
                              void* d_out, int out_size, void* d_ws, size_t ws_size,
                              hipStream_t stream) {
    const float* action_logps = (const float*)d_in[0];  // (T+1, 128, 256)
    const float* stop_logps   = (const float*)d_in[1];  // (T+1, 128, 2)
    const float* start_logps  = (const float*)d_in[2];  // (T+1, 128)
    const int*   actions      = (const int*)d_in[3];    // (T,)
    const int T = in_sizes[3];

    float* al = (float*)d_ws;                           // (T, 128) = 4 MB

    const int n = T * NB;
    gather_al<<<(n + 255) / 256, 256, 0, stream>>>(action_logps, actions, al, T);
    hmm_scan<<<1, 32, 0, stream>>>(stop_logps, start_logps, al, (float*)d_out, T);
}